// ECNConvNN_14671608283725
// MI455X (gfx1250) — compile-verified
//
#include <hip/hip_runtime.h>

// CDNA5 (gfx1250) NNConv via V_WMMA_F32_16X16X4_F32.
//
// msg[e,o] = sum_f sum_i (ea[e,f] * x[src[e],i]) * w_nn[f, i*16+o]  + (x[src] @ Bnn)
// => per 16-edge tile: one 16x128x16 fp32 GEMM (32 chained wmma 16x16x4)
//    + 4 wmma steps for the b_nn term. Root term is a separate WMMA kernel.

typedef __attribute__((ext_vector_type(2))) float v2f;
typedef __attribute__((ext_vector_type(4))) float v4f;
typedef __attribute__((ext_vector_type(8))) float v8f;
typedef __attribute__((ext_vector_type(4))) int   v4i;

__device__ __forceinline__ v8f wmma4(v2f a, v2f b, v8f c) {
  // D = A(16x4,f32) * B(4x16,f32) + C(16x16,f32)
  return __builtin_amdgcn_wmma_f32_16x16x4_f32(
      /*neg_a=*/false, a, /*neg_b=*/false, b,
      /*c_mod=*/(short)0, c, /*reuse_a=*/false, /*reuse_b=*/false);
}

// -------------------------- edge (message + scatter) kernel -----------------
__global__ void __launch_bounds__(256)
ecn_edge_kernel(const float* __restrict__ x,      // [N,16]
                const int*   __restrict__ eidx,   // [2,E] (src row, dst row)
                const float* __restrict__ eattr,  // [E,8]
                const float* __restrict__ w_nn,   // [8,256]
                const float* __restrict__ b_nn,   // [256]
                float*       __restrict__ out,    // [N,16] pre-init x@root+bias
                int nEdges)
{
  const int lane = threadIdx.x & 31;
  const int half = lane >> 4;   // 0: lanes 0-15, 1: lanes 16-31
  const int lm   = lane & 15;
  const int waveId = (blockIdx.x * blockDim.x + threadIdx.x) >> 5;
  const int nWaves = (gridDim.x * blockDim.x) >> 5;
  const int nTiles = (nEdges + 15) >> 4;
  const int* __restrict__ srcp = eidx;
  const int* __restrict__ dstp = eidx + nEdges;

  // Constant B operand: w_nn viewed as [K=128 (f*16+i), N=16], chunked into
  // 32 (4xK)x16 slabs. Lane L, vgpr v holds B[k0 + v + 2*half][lm].
  v2f bW[32];
#pragma unroll
  for (int c = 0; c < 32; ++c) {
    const int f = c >> 2;                       // compile-time: K never crosses f within a chunk
#pragma unroll
    for (int v = 0; v < 2; ++v) {
      const int i = 4 * (c & 3) + v + 2 * half; // 0..15 within f
      bW[c][v] = w_nn[f * 256 + i * 16 + lm];
    }
  }
  // b_nn term: B[i][o] = b_nn[i*16+o], 4 chunks.
  v2f bB[4];
#pragma unroll
  for (int c = 0; c < 4; ++c)
#pragma unroll
    for (int v = 0; v < 2; ++v)
      bB[c][v] = b_nn[(4 * c + v + 2 * half) * 16 + lm];

  for (int t = waveId; t < nTiles; t += nWaves) {
    const int  e0  = t << 4;
    const int  em  = e0 + lm;
    const bool vld = (em < nEdges);
    const int  es  = vld ? em : 0;
    const int  s   = srcp[es];

    // Per-lane slice of x[src]: this lane only needs i = 4g + {0,1} + 2*half.
    // xs[2g+w] = x[s][4g + w + 2*half]  (lanes L and L+16 cover disjoint halves)
    float xs[8];
#pragma unroll
    for (int g = 0; g < 4; ++g) {
      v2f xv = *(const v2f*)(x + (size_t)s * 16 + 4 * g + 2 * half);
      xs[2 * g + 0] = xv[0];
      xs[2 * g + 1] = xv[1];
    }
    float ear[8];
    {
      const v4f* ep = (const v4f*)(eattr + (size_t)es * 8);
      v4f a0 = ep[0], a1 = ep[1];
      ear[0] = a0.x; ear[1] = a0.y; ear[2] = a0.z; ear[3] = a0.w;
      ear[4] = a1.x; ear[5] = a1.y; ear[6] = a1.z; ear[7] = a1.w;
    }
    if (!vld) {
#pragma unroll
      for (int q = 0; q < 8; ++q) { xs[q] = 0.f; ear[q] = 0.f; }
    }

    // dst rows this lane scatters: M = v + 8*half  ->  dst[e0 + 8*half + v]
    int dsts[8];
    const int dbase = e0 + 8 * half;
    if (e0 + 16 <= nEdges) {                    // wave-uniform branch
      v4i d0 = *(const v4i*)(dstp + dbase);
      v4i d1 = *(const v4i*)(dstp + dbase + 4);
      dsts[0] = d0.x; dsts[1] = d0.y; dsts[2] = d0.z; dsts[3] = d0.w;
      dsts[4] = d1.x; dsts[5] = d1.y; dsts[6] = d1.z; dsts[7] = d1.w;
    } else {
#pragma unroll
      for (int v = 0; v < 8; ++v) {
        const int ei = dbase + v;
        dsts[v] = (ei < nEdges) ? dstp[ei] : -1;
      }
    }

    // 16x128x16 GEMM: A'[e, f*16+i] = ea[e,f] * x_src[e,i]
    v8f acc = {0.f, 0.f, 0.f, 0.f, 0.f, 0.f, 0.f, 0.f};
#pragma unroll
    for (int c = 0; c < 32; ++c) {
      const int f = c >> 2;
      v2f a;
      a[0] = ear[f] * xs[2 * (c & 3) + 0];
      a[1] = ear[f] * xs[2 * (c & 3) + 1];
      acc = wmma4(a, bW[c], acc);
    }
    // + x_src @ reshape(b_nn, 16x16)
#pragma unroll
    for (int c = 0; c < 4; ++c) {
      v2f a;
      a[0] = xs[2 * c + 0];
      a[1] = xs[2 * c + 1];
      acc = wmma4(a, bB[c], acc);
    }

    // Scatter-add: acc[v] is msg[row M = v + 8*half, col lm]
#pragma unroll
    for (int v = 0; v < 8; ++v) {
      const int d = dsts[v];
      if (d >= 0)
        unsafeAtomicAdd(out + (size_t)d * 16 + lm, acc[v]);
    }
  }
}

// -------------------------- node (root + bias init) kernel ------------------
__global__ void __launch_bounds__(256)
ecn_node_kernel(const float* __restrict__ x,     // [N,16]
                const float* __restrict__ root,  // [16,16]
                const float* __restrict__ bias,  // [16]
                float*       __restrict__ out,   // [N,16]
                int nNodes)
{
  const int lane = threadIdx.x & 31;
  const int half = lane >> 4;
  const int lm   = lane & 15;
  const int waveId = (blockIdx.x * blockDim.x + threadIdx.x) >> 5;
  const int nTiles = (nNodes + 15) >> 4;
  if (waveId >= nTiles) return;                 // wave-uniform: EXEC stays all-1s
  const int n0 = waveId << 4;

  v2f bR[4];
#pragma unroll
  for (int c = 0; c < 4; ++c)
#pragma unroll
    for (int v = 0; v < 2; ++v)
      bR[c][v] = root[(4 * c + v + 2 * half) * 16 + lm];

  const int nr = n0 + lm;
  const int ns = (nr < nNodes) ? nr : (nNodes - 1);
  float xs[8];
#pragma unroll
  for (int g = 0; g < 4; ++g) {
    v2f xv = *(const v2f*)(x + (size_t)ns * 16 + 4 * g + 2 * half);
    xs[2 * g + 0] = xv[0];
    xs[2 * g + 1] = xv[1];
  }

  const float bv = bias[lm];                    // C[M][N] = bias[N]
  v8f acc = {bv, bv, bv, bv, bv, bv, bv, bv};
#pragma unroll
  for (int c = 0; c < 4; ++c) {
    v2f a;
    a[0] = xs[2 * c + 0];
    a[1] = xs[2 * c + 1];
    acc = wmma4(a, bR[c], acc);
  }

#pragma unroll
  for (int v = 0; v < 8; ++v) {
    const int m = n0 + v + 8 * half;
    if (m < nNodes)
      out[(size_t)m * 16 + lm] = acc[v];
  }
}

// ---------------------------------------------------------------------------
extern "C" void kernel_launch(void* const* d_in, const int* in_sizes, int n_in,
                              void* d_out, int out_size, void* d_ws, size_t ws_size,
                              hipStream_t stream) {
  const float* x    = (const float*)d_in[0];
  const int*   eidx = (const int*)  d_in[1];
  const float* ea   = (const float*)d_in[2];
  const float* wnn  = (const float*)d_in[3];
  const float* bnn  = (const float*)d_in[4];
  const float* root = (const float*)d_in[5];
  const float* bias = (const float*)d_in[6];
  float* out = (float*)d_out;

  const int nNodes = in_sizes[0] / 16;   // 50000
  const int nEdges = in_sizes[2] / 8;    // 800000

  // Pass-through tuple outputs: edge_index (int bits) then edge_attr.
  const size_t off1 = (size_t)nNodes * 16;
  hipMemcpyAsync(out + off1, eidx, (size_t)2 * nEdges * sizeof(int),
                 hipMemcpyDeviceToDevice, stream);
  hipMemcpyAsync(out + off1 + (size_t)2 * nEdges, ea,
                 (size_t)nEdges * 8 * sizeof(float),
                 hipMemcpyDeviceToDevice, stream);

  // 1) out = x @ root + bias   (must precede edge scatter-adds; stream-ordered)
  const int nodeTiles  = (nNodes + 15) / 16;
  const int nodeBlocks = (nodeTiles + 7) / 8;   // 8 waves per 256-thread block
  ecn_node_kernel<<<nodeBlocks, 256, 0, stream>>>(x, root, bias, out, nNodes);

  // 2) out += scatter_add(messages)
  ecn_edge_kernel<<<512, 256, 0, stream>>>(x, eidx, ea, wnn, bnn, out, nEdges);
}